// GCNWithEdge_33139967656166
// MI455X (gfx1250) — compile-verified
//
#include <hip/hip_runtime.h>
#include <hip/hip_bf16.h>

#define N_NODES_C 100000
#define N_EDGES_C 1600000

typedef __attribute__((ext_vector_type(2))) float v2f;
typedef __attribute__((ext_vector_type(8))) float v8f;

// ---------------------------------------------------------------------------
// Degree accumulation: deg[d] += 1 for each edge (self-loops folded in later)
// ---------------------------------------------------------------------------
__global__ void deg_kernel(const int* __restrict__ dst, float* __restrict__ deg, int nE) {
    int e = blockIdx.x * blockDim.x + threadIdx.x;
    if (e < nE) atomicAdd(&deg[dst[e]], 1.0f);
}

// dinv[i] = rsqrt(deg[i] + 1)   (+1 = self loop; always >= 1)
__global__ void dinv_kernel(float* __restrict__ deg, int n) {
    int i = blockIdx.x * blockDim.x + threadIdx.x;
    if (i < n) deg[i] = rsqrtf(deg[i] + 1.0f);
}

// ---------------------------------------------------------------------------
// C[M x 64] = A[M x 64] @ W[64 x 64]  via V_WMMA_F32_16X16X4_F32
// One wave per 16-row strip; 4 accumulators cover the 64 output columns;
// K=64 in 16 steps of 4. W staged in LDS.
//
// A 16x4 f32 layout (ISA 7.12.2): lane L holds M=L%16, VGPR j holds
// K = 4*kk + 2*(L/16) + j.  B 4x16: VGPR j, lane L -> K = 2*(L/16)+j, N=L%16.
// C 16x16: VGPR r, lane L -> M = r + 8*(L/16), N = L%16.
// ---------------------------------------------------------------------------
__global__ void __launch_bounds__(128)
gemm_n64_wmma(const float* __restrict__ A, const float* __restrict__ W,
              float* __restrict__ C, int M) {
    __shared__ float Wl[64 * 64];
    for (int i = threadIdx.x; i < 64 * 64; i += blockDim.x) Wl[i] = W[i];
    __syncthreads();

    const int wave = threadIdx.x >> 5;
    const int lane = threadIdx.x & 31;
    const int strip = blockIdx.x * 4 + wave;          // 16-row strip index
    if (strip * 16 >= M) return;                      // wave-uniform guard (EXEC stays all-1)

    const int mlo  = lane & 15;
    const int half = lane >> 4;                       // 0 or 1
    const float* __restrict__ arow = A + (size_t)(strip * 16 + mlo) * 64;

    v8f acc0 = {}, acc1 = {}, acc2 = {}, acc3 = {};
    #pragma unroll
    for (int kk = 0; kk < 16; ++kk) {
        const int k0 = kk * 4 + half * 2;
        v2f a;
        a.x = arow[k0];
        a.y = arow[k0 + 1];
        v2f b0, b1, b2, b3;
        b0.x = Wl[(k0    ) * 64 +  0 + mlo];  b0.y = Wl[(k0 + 1) * 64 +  0 + mlo];
        b1.x = Wl[(k0    ) * 64 + 16 + mlo];  b1.y = Wl[(k0 + 1) * 64 + 16 + mlo];
        b2.x = Wl[(k0    ) * 64 + 32 + mlo];  b2.y = Wl[(k0 + 1) * 64 + 32 + mlo];
        b3.x = Wl[(k0    ) * 64 + 48 + mlo];  b3.y = Wl[(k0 + 1) * 64 + 48 + mlo];
        acc0 = __builtin_amdgcn_wmma_f32_16x16x4_f32(false, a, false, b0, (short)0, acc0, false, false);
        acc1 = __builtin_amdgcn_wmma_f32_16x16x4_f32(false, a, false, b1, (short)0, acc1, false, false);
        acc2 = __builtin_amdgcn_wmma_f32_16x16x4_f32(false, a, false, b2, (short)0, acc2, false, false);
        acc3 = __builtin_amdgcn_wmma_f32_16x16x4_f32(false, a, false, b3, (short)0, acc3, false, false);
    }

    #pragma unroll
    for (int r = 0; r < 8; ++r) {
        const size_t m = (size_t)(strip * 16 + r + 8 * half);
        float* crow = C + m * 64 + mlo;
        crow[ 0] = acc0[r];
        crow[16] = acc1[r];
        crow[32] = acc2[r];
        crow[48] = acc3[r];
    }
}

// ---------------------------------------------------------------------------
// Edge scatter: agg[d] += h[s] * dinv[s]*dinv[d].  16 threads/edge, float4.
// ---------------------------------------------------------------------------
__global__ void scatter_edges(const float* __restrict__ h, const int* __restrict__ src,
                              const int* __restrict__ dst, const float* __restrict__ dinv,
                              float* __restrict__ agg, int nE) {
    int t = blockIdx.x * blockDim.x + threadIdx.x;
    int e = t >> 4;
    if (e >= nE) return;
    int q = t & 15;
    int s = src[e], d = dst[e];
    float w = dinv[s] * dinv[d];
    const float4* hv = (const float4*)(h + (size_t)s * 64);
    float4 v = hv[q];
    float* p = agg + (size_t)d * 64 + q * 4;
    atomicAdd(p + 0, v.x * w);
    atomicAdd(p + 1, v.y * w);
    atomicAdd(p + 2, v.z * w);
    atomicAdd(p + 3, v.w * w);
}

// ---------------------------------------------------------------------------
// Self-loop + bias (+ optional ReLU):  agg[i][f] += h[i][f]*dinv[i]^2 + b[f]
// ---------------------------------------------------------------------------
__global__ void finish_nodes(const float* __restrict__ h, const float* __restrict__ dinv,
                             const float* __restrict__ b, float* __restrict__ agg,
                             int nN, int doRelu) {
    int t = blockIdx.x * blockDim.x + threadIdx.x;
    int i = t >> 6;
    if (i >= nN) return;
    int f = t & 63;
    float di = dinv[i];
    float v = agg[(size_t)i * 64 + f] + h[(size_t)i * 64 + f] * di * di + b[f];
    if (doRelu) v = fmaxf(v, 0.0f);
    agg[(size_t)i * 64 + f] = v;
}

// ---------------------------------------------------------------------------
// Fused edge MLP: [h[s]-h[d] | ea0,ea1 | emb0[ea2] | emb1[ea3]] (130)
//   -> fcW1 (130x2) + fcb1 -> relu -> fcW2 (2x2) + fcb2 -> log_softmax
// ---------------------------------------------------------------------------
__global__ void edge_mlp(const float* __restrict__ h, const int* __restrict__ src,
                         const int* __restrict__ dst, const int* __restrict__ ea,
                         const float* __restrict__ emb0, const float* __restrict__ emb1,
                         const float* __restrict__ fcW1, const float* __restrict__ fcb1,
                         const float* __restrict__ fcW2, const float* __restrict__ fcb2,
                         float* __restrict__ out, int nE) {
    int e = blockIdx.x * blockDim.x + threadIdx.x;
    if (e >= nE) return;
    int s = src[e], d = dst[e];
    float z0 = fcb1[0], z1 = fcb1[1];

    const float4* hs = (const float4*)(h + (size_t)s * 64);
    const float4* hd = (const float4*)(h + (size_t)d * 64);
    #pragma unroll
    for (int q = 0; q < 16; ++q) {
        float4 a = hs[q], b = hd[q];
        int f = q * 4;
        float t0 = a.x - b.x, t1 = a.y - b.y, t2 = a.z - b.z, t3 = a.w - b.w;
        z0 += t0 * fcW1[(f + 0) * 2 + 0] + t1 * fcW1[(f + 1) * 2 + 0]
            + t2 * fcW1[(f + 2) * 2 + 0] + t3 * fcW1[(f + 3) * 2 + 0];
        z1 += t0 * fcW1[(f + 0) * 2 + 1] + t1 * fcW1[(f + 1) * 2 + 1]
            + t2 * fcW1[(f + 2) * 2 + 1] + t3 * fcW1[(f + 3) * 2 + 1];
    }

    float v0 = (float)ea[e];
    float v1 = (float)ea[(size_t)nE + e];
    z0 += v0 * fcW1[64 * 2 + 0] + v1 * fcW1[65 * 2 + 0];
    z1 += v0 * fcW1[64 * 2 + 1] + v1 * fcW1[65 * 2 + 1];

    int i2 = ea[(size_t)2 * nE + e];
    int i3 = ea[(size_t)3 * nE + e];
    const float* e0 = emb0 + (size_t)i2 * 32;
    const float* e1 = emb1 + (size_t)i3 * 32;
    #pragma unroll
    for (int j = 0; j < 32; ++j) {
        z0 += e0[j] * fcW1[(66 + j) * 2 + 0] + e1[j] * fcW1[(98 + j) * 2 + 0];
        z1 += e0[j] * fcW1[(66 + j) * 2 + 1] + e1[j] * fcW1[(98 + j) * 2 + 1];
    }

    z0 = fmaxf(z0, 0.0f);
    z1 = fmaxf(z1, 0.0f);
    float o0 = z0 * fcW2[0] + z1 * fcW2[2] + fcb2[0];
    float o1 = z0 * fcW2[1] + z1 * fcW2[3] + fcb2[1];
    float m  = fmaxf(o0, o1);
    float l  = m + logf(expf(o0 - m) + expf(o1 - m));
    out[(size_t)e * 2 + 0] = o0 - l;
    out[(size_t)e * 2 + 1] = o1 - l;
}

// ---------------------------------------------------------------------------
extern "C" void kernel_launch(void* const* d_in, const int* in_sizes, int n_in,
                              void* d_out, int out_size, void* d_ws, size_t ws_size,
                              hipStream_t stream) {
    const float* x    = (const float*)d_in[0];
    const int*   ei   = (const int*)d_in[1];
    const int*   ea   = (const int*)d_in[2];
    const float* W1   = (const float*)d_in[3];
    const float* b1   = (const float*)d_in[4];
    const float* W2   = (const float*)d_in[5];
    const float* b2   = (const float*)d_in[6];
    const float* emb0 = (const float*)d_in[7];
    const float* emb1 = (const float*)d_in[8];
    const float* fcW1 = (const float*)d_in[9];
    const float* fcb1 = (const float*)d_in[10];
    const float* fcW2 = (const float*)d_in[11];
    const float* fcb2 = (const float*)d_in[12];
    float* out = (float*)d_out;

    const int nN = in_sizes[0] / 64;   // 100000
    const int nE = in_sizes[1] / 2;    // 1600000
    const int* srcI = ei;
    const int* dstI = ei + nE;

    float* dinv = (float*)d_ws;                       // nN floats (400000 B, 16B aligned)
    float* bufA = dinv + nN;                          // nN*64 floats
    float* bufB = bufA + (size_t)nN * 64;             // nN*64 floats

    const size_t nodeFeatBytes = (size_t)nN * 64 * sizeof(float);
    const int strips = (nN + 15) / 16;
    const int gemmBlocks = (strips + 3) / 4;

    // degrees -> dinv
    hipMemsetAsync(dinv, 0, (size_t)nN * sizeof(float), stream);
    deg_kernel<<<(nE + 255) / 256, 256, 0, stream>>>(dstI, dinv, nE);
    dinv_kernel<<<(nN + 255) / 256, 256, 0, stream>>>(dinv, nN);

    // ---- layer 1 ----
    gemm_n64_wmma<<<gemmBlocks, 128, 0, stream>>>(x, W1, bufA, nN);
    hipMemsetAsync(bufB, 0, nodeFeatBytes, stream);
    {
        long long tt = (long long)nE * 16;
        scatter_edges<<<(int)((tt + 255) / 256), 256, 0, stream>>>(bufA, srcI, dstI, dinv, bufB, nE);
    }
    {
        long long tt = (long long)nN * 64;
        finish_nodes<<<(int)((tt + 255) / 256), 256, 0, stream>>>(bufA, dinv, b1, bufB, nN, 1);
    }

    // ---- layer 2 ----
    gemm_n64_wmma<<<gemmBlocks, 128, 0, stream>>>(bufB, W2, bufA, nN);
    hipMemsetAsync(bufB, 0, nodeFeatBytes, stream);
    {
        long long tt = (long long)nE * 16;
        scatter_edges<<<(int)((tt + 255) / 256), 256, 0, stream>>>(bufA, srcI, dstI, dinv, bufB, nE);
    }
    {
        long long tt = (long long)nN * 64;
        finish_nodes<<<(int)((tt + 255) / 256), 256, 0, stream>>>(bufA, dinv, b2, bufB, nN, 0);
    }

    // ---- fused edge MLP + log_softmax ----
    edge_mlp<<<(nE + 255) / 256, 256, 0, stream>>>(bufB, srcI, dstI, ea, emb0, emb1,
                                                   fcW1, fcb1, fcW2, fcb2, out, nE);
}